// GCNModel_79559974191165
// MI455X (gfx1250) — compile-verified
//
#include <hip/hip_runtime.h>

typedef float v2f __attribute__((ext_vector_type(2)));
typedef float v8f __attribute__((ext_vector_type(8)));

#define HFEAT 64
#define BSTRIDE 80   // LDS row stride in v2f units; 160 dwords = 32 mod 64 banks

// -------------------------------------------------------------------------
// degree init: every node gets a self-loop of weight 1
__global__ void deg_init_kernel(float* __restrict__ deg, int n) {
    int i = blockIdx.x * blockDim.x + threadIdx.x;
    if (i < n) deg[i] = 1.0f;
}

// scatter edge weights into degree at dst
__global__ void deg_scatter_kernel(const long long* __restrict__ dst,
                                   const float* __restrict__ ew,
                                   float* __restrict__ deg, long long nedges) {
    long long e = (long long)blockIdx.x * blockDim.x + threadIdx.x;
    if (e < nedges) atomicAdd(&deg[(int)dst[e]], ew[e]);
}

// deg -> deg^{-1/2} in place
__global__ void dis_kernel(float* __restrict__ deg, int n) {
    int i = blockIdx.x * blockDim.x + threadIdx.x;
    if (i < n) {
        float d = deg[i];
        deg[i] = (d > 0.0f) ? rsqrtf(fmaxf(d, 1e-12f)) : 0.0f;
    }
}

// per-edge symmetric norm
__global__ void enorm_kernel(const long long* __restrict__ src,
                             const long long* __restrict__ dst,
                             const float* __restrict__ ew,
                             const float* __restrict__ dis,
                             float* __restrict__ enorm, long long nedges) {
    long long e = (long long)blockIdx.x * blockDim.x + threadIdx.x;
    if (e < nedges) enorm[e] = dis[(int)src[e]] * ew[e] * dis[(int)dst[e]];
}

// -------------------------------------------------------------------------
// C[n,64] = A[n,K] @ B[K,64] with V_WMMA_F32_16X16X4_F32.
// One wave -> 16 rows x 64 cols (4 accumulators). 8 waves/block -> 128 rows.
// B staged in LDS pre-paired as (B[2j][col], B[2j+1][col]) so each fragment
// is one ds_load_b64; row stride padded for bank-conflict-free half-waves.
template <int K>
__global__ void gemm_wmma_f32_kernel(const float* __restrict__ A,
                                     const float* __restrict__ B,
                                     float* __restrict__ C, int nrows) {
    __shared__ v2f Bs[(K / 2) * BSTRIDE];
    for (int i = threadIdx.x; i < (K / 2) * HFEAT; i += blockDim.x) {
        int j = i / HFEAT;
        int col = i - j * HFEAT;
        v2f p;
        p.x = B[(2 * j + 0) * HFEAT + col];
        p.y = B[(2 * j + 1) * HFEAT + col];
        Bs[j * BSTRIDE + col] = p;
    }
    __syncthreads();

    const int wave = threadIdx.x >> 5;
    const int lane = threadIdx.x & 31;
    const int g = lane >> 4;      // lane-half selector (K pair / row-half)
    const int nn = lane & 15;     // column / row-in-tile index

    const int row0 = blockIdx.x * 128 + wave * 16;
    int rowa = row0 + nn;
    if (rowa >= nrows) rowa = nrows - 1;          // clamp loads, keep EXEC full
    const float* arow = A + (size_t)rowa * K;

    v8f zero;
#pragma unroll
    for (int r = 0; r < 8; ++r) zero[r] = 0.0f;
    v8f acc[4] = {zero, zero, zero, zero};

    for (int k = 0; k < K; k += 4) {
        // A fragment (16x4): lane (g,nn) holds A[row][k+2g], A[row][k+2g+1]
        v2f a = *(const v2f*)(arow + k + 2 * g);
        const v2f* brow = Bs + ((k >> 1) + g) * BSTRIDE + nn;
#pragma unroll
        for (int t = 0; t < 4; ++t) {
            v2f b = brow[t * 16];   // (B[k+2g][col], B[k+2g+1][col])
            acc[t] = __builtin_amdgcn_wmma_f32_16x16x4_f32(
                false, a, false, b, (short)0, acc[t], false, false);
        }
    }

    // D layout: VGPR r -> M = row0 + 8*g + r, N = nn + 16*t
    float* crow = C + (size_t)(row0 + 8 * g) * HFEAT + nn;
    if (row0 + 16 <= nrows) {
        // fast path: whole 16-row slab in range, unguarded coalesced stores
#pragma unroll
        for (int r = 0; r < 8; ++r) {
#pragma unroll
            for (int t = 0; t < 4; ++t) {
                crow[(size_t)r * HFEAT + t * 16] = acc[t][r];
            }
        }
    } else {
        // tail block only
#pragma unroll
        for (int r = 0; r < 8; ++r) {
            int m = row0 + 8 * g + r;
            if (m < nrows) {
#pragma unroll
                for (int t = 0; t < 4; ++t) {
                    crow[(size_t)r * HFEAT + t * 16] = acc[t][r];
                }
            }
        }
    }
}

// -------------------------------------------------------------------------
// out[i][f] = hw[i][f] * dis[i]^2 + bias[f]   (self-loop contribution + bias)
__global__ void agg_init_kernel(const float* __restrict__ hw,
                                const float* __restrict__ dis,
                                const float* __restrict__ bias,
                                float* __restrict__ out, int n) {
    long long tid = (long long)blockIdx.x * blockDim.x + threadIdx.x;
    int i = (int)(tid >> 4);
    int f4 = (int)(tid & 15);
    if (i >= n) return;
    float s = dis[i];
    s = s * s;
    float4 v = ((const float4*)(hw + (size_t)i * HFEAT))[f4];
    float4 b = ((const float4*)bias)[f4];
    float4 o;
    o.x = v.x * s + b.x;
    o.y = v.y * s + b.y;
    o.z = v.z * s + b.z;
    o.w = v.w * s + b.w;
    ((float4*)(out + (size_t)i * HFEAT))[f4] = o;
}

// scatter: out[dst] += hw[src] * enorm[e], 16 threads/edge, 4 feats each
__global__ void agg_edges_kernel(const float* __restrict__ hw,
                                 const long long* __restrict__ src,
                                 const long long* __restrict__ dst,
                                 const float* __restrict__ enorm,
                                 float* __restrict__ out, long long nedges) {
    long long tid = (long long)blockIdx.x * blockDim.x + threadIdx.x;
    long long e = tid >> 4;
    int f4 = (int)(tid & 15);
    if (e >= nedges) return;
    int s = (int)src[e];
    int d = (int)dst[e];
    float w = enorm[e];
    float4 v = ((const float4*)(hw + (size_t)s * HFEAT))[f4];
    float* o = out + (size_t)d * HFEAT + f4 * 4;
    atomicAdd(o + 0, v.x * w);
    atomicAdd(o + 1, v.y * w);
    atomicAdd(o + 2, v.z * w);
    atomicAdd(o + 3, v.w * w);
}

// -------------------------------------------------------------------------
// out[i] = round(clip(h[i] . fc_w + fc_b, 0, 10))
__global__ void head_kernel(const float* __restrict__ h,
                            const float* __restrict__ fcw,
                            const float* __restrict__ fcb,
                            float* __restrict__ out, int n) {
    int i = blockIdx.x * blockDim.x + threadIdx.x;
    if (i >= n) return;
    const float4* row = (const float4*)(h + (size_t)i * HFEAT);
    const float4* w4 = (const float4*)fcw;
    float acc = 0.0f;
#pragma unroll
    for (int f = 0; f < HFEAT / 4; ++f) {
        float4 v = row[f];
        float4 w = w4[f];
        acc += v.x * w.x + v.y * w.y + v.z * w.z + v.w * w.w;
    }
    acc += fcb[0];
    acc = fminf(fmaxf(acc, 0.0f), 10.0f);
    out[i] = rintf(acc);   // round-to-nearest-even, matches jnp.round
}

// -------------------------------------------------------------------------
extern "C" void kernel_launch(void* const* d_in, const int* in_sizes, int n_in,
                              void* d_out, int out_size, void* d_ws, size_t ws_size,
                              hipStream_t stream) {
    const float* x       = (const float*)d_in[0];
    const long long* ei  = (const long long*)d_in[1];
    const float* ew      = (const float*)d_in[2];
    const float* W1      = (const float*)d_in[3];
    const float* b1      = (const float*)d_in[4];
    const float* W2      = (const float*)d_in[5];
    const float* b2      = (const float*)d_in[6];
    const float* fcw     = (const float*)d_in[7];
    const float* fcb     = (const float*)d_in[8];
    float* out           = (float*)d_out;

    const int DIN = 128;
    const int n = in_sizes[0] / DIN;          // 100000
    const long long E = in_sizes[1] / 2;      // 1600000
    const long long* src = ei;
    const long long* dst = ei + E;

    // carve workspace
    char* ws = (char*)d_ws;
    size_t off = 0;
    auto carve = [&](size_t bytes) -> void* {
        off = (off + 255) & ~(size_t)255;
        void* p = ws + off;
        off += bytes;
        return p;
    };
    float* dis   = (float*)carve((size_t)n * sizeof(float));
    float* enorm = (float*)carve((size_t)E * sizeof(float));
    float* bufA  = (float*)carve((size_t)n * HFEAT * sizeof(float)); // hw
    float* bufB  = (float*)carve((size_t)n * HFEAT * sizeof(float)); // h
    (void)ws_size;

    const int TB = 256;
    int nB   = (n + TB - 1) / TB;
    int eB   = (int)((E + TB - 1) / TB);
    int n16B = (int)(((long long)n * 16 + TB - 1) / TB);
    int e16B = (int)((E * 16 + TB - 1) / TB);
    int gemmB = (n + 127) / 128;

    // ---- normalization ----
    deg_init_kernel<<<nB, TB, 0, stream>>>(dis, n);
    deg_scatter_kernel<<<eB, TB, 0, stream>>>(dst, ew, dis, E);
    dis_kernel<<<nB, TB, 0, stream>>>(dis, n);
    enorm_kernel<<<eB, TB, 0, stream>>>(src, dst, ew, dis, enorm, E);

    // ---- layer 1: hw = x@W1 ; h1 = scatter(hw*norm) + self + b1 ----
    gemm_wmma_f32_kernel<128><<<gemmB, TB, 0, stream>>>(x, W1, bufA, n);
    agg_init_kernel<<<n16B, TB, 0, stream>>>(bufA, dis, b1, bufB, n);
    agg_edges_kernel<<<e16B, TB, 0, stream>>>(bufA, src, dst, enorm, bufB, E);

    // ---- layer 2: hw2 = h1@W2 ; h2 = scatter(hw2*norm) + self + b2 ----
    gemm_wmma_f32_kernel<64><<<gemmB, TB, 0, stream>>>(bufB, W2, bufA, n);
    agg_init_kernel<<<n16B, TB, 0, stream>>>(bufA, dis, b2, bufB, n);
    agg_edges_kernel<<<e16B, TB, 0, stream>>>(bufA, src, dst, enorm, bufB, E);

    // ---- head ----
    head_kernel<<<nB, TB, 0, stream>>>(bufB, fcw, fcb, out, n);
}